// CrossEntropyLoss_71133248356852
// MI455X (gfx1250) — compile-verified
//
#include <hip/hip_runtime.h>
#include <stdint.h>

// ---------------- problem constants (fixed by the reference) ----------------
#define C_CLS           10          // classes per row
#define NEXP            9           // exp poly coeffs (deg 8)
#define NINV            5           // inverse poly coeffs (deg 4)
#define NLOG            9           // log poly coeffs (deg 8)

// ---------------- tiling ----------------
#define ROWS_PER_TILE   250
#define TILE_ELEMS      (ROWS_PER_TILE * C_CLS)   // 2500 floats = 10000 B
#define TILES_PER_BLOCK 4
#define NT_MAIN         256

__device__ __forceinline__ uint32_t rfl(uint32_t v) {
  return (uint32_t)__builtin_amdgcn_readfirstlane((int)v);
}

// Build a Tensor-DMA descriptor (D#) for a 1-D contiguous global->LDS copy and
// issue TENSOR_LOAD_TO_LDS. Layout per CDNA5 ISA ch.8:
//   group0: [1:0]=count=1, [63:32]=lds_addr, [120:64]=global_addr, [127:126]=type=2
//   group1: [17:16]=data_size(2 => 4B), [79:48]=tensor_dim0, [111:80]=tensor_dim1,
//           [127:112]=tile_dim0, [143:128]=tile_dim1(0=unused),
//           [207:160]=tensor_dim0_stride, [255:208]=tensor_dim1_stride
// VADDR2/VADDR3 omitted (2-group form, tensors up to 2D).
// Every word goes through readfirstlane (values are wave-uniform) so the
// copies into the pinned SGPRs are legal s_mov_b32. Pins live at the TOP of
// the SGPR file (s84..s95) so they don't collide with the allocator's hot
// low range (previous pin at s36..s47 caused SGPR->VGPR-lane spills).
__device__ __forceinline__ void tdm_load_1d(uint32_t lds_off, const float* gsrc,
                                            uint32_t nelem) {
  uint64_t ga = (uint64_t)(uintptr_t)gsrc;
  const uint32_t d0 = rfl(1u);                                   // count=1
  const uint32_t d1 = rfl(lds_off);                              // LDS byte addr
  const uint32_t d2 = rfl((uint32_t)ga);                         // gaddr[31:0]
  const uint32_t d3 = rfl(((uint32_t)(ga >> 32) & 0x01FFFFFFu)   // gaddr[56:32]
                          | (2u << 30));                         // type=2
  const uint32_t e0 = rfl(2u << 16);                             // data_size=4B
  const uint32_t e1 = rfl((nelem & 0xFFFFu) << 16);              // tdim0[15:0]
  const uint32_t e2 = rfl(((nelem >> 16) & 0xFFFFu)              // tdim0[31:16]
                          | (1u << 16));                         // tdim1=1
  const uint32_t e3 = rfl((nelem & 0xFFFFu) << 16);              // tile_dim0
  const uint32_t e4 = rfl(0u);                                   // tile_dim1/2=0
  const uint32_t e5 = rfl(nelem);                                // stride0[31:0]
  const uint32_t e6 = rfl((nelem & 0xFFFFu) << 16);              // stride1[15:0]
  const uint32_t e7 = rfl(nelem >> 16);                          // stride1[47:16]
  asm volatile("tensor_load_to_lds s[84:87], s[88:95]"
               :
               : "{s84}"(d0), "{s85}"(d1), "{s86}"(d2), "{s87}"(d3),
                 "{s88}"(e0), "{s89}"(e1), "{s90}"(e2), "{s91}"(e3),
                 "{s92}"(e4), "{s93}"(e5), "{s94}"(e6), "{s95}"(e7)
               : "memory");
}

// float2 helpers — pairwise fp32 math so the backend can pack v_pk_fma_f32
__device__ __forceinline__ float2 f2_fma(float2 a, float2 b, float2 c) {
  return make_float2(fmaf(a.x, b.x, c.x), fmaf(a.y, b.y, c.y));
}
__device__ __forceinline__ float2 f2_bcast(float v) { return make_float2(v, v); }
__device__ __forceinline__ float2 f2_mul(float2 a, float2 b) {
  return make_float2(a.x * b.x, a.y * b.y);
}

__global__ __launch_bounds__(NT_MAIN)
void ce_poly_main(const float* __restrict__ xg, const float* __restrict__ tg,
                  const float* __restrict__ ec_g, const float* __restrict__ ic_g,
                  const float* __restrict__ lc_g, const int* __restrict__ it_g,
                  float* __restrict__ partials, int rows_total, int ntiles_total) {
  __shared__ float xbuf[2][TILE_ELEMS];
  __shared__ float tbuf[2][TILE_ELEMS];
  __shared__ float wred[NT_MAIN / 32];

  // Coefficients: uniform, tiny; keep in registers with constant indices.
  float ec[NEXP], ic[NINV], lc[NLOG];
#pragma unroll
  for (int i = 0; i < NEXP; ++i) ec[i] = ec_g[i];
#pragma unroll
  for (int i = 0; i < NINV; ++i) ic[i] = ic_g[i];
#pragma unroll
  for (int i = 0; i < NLOG; ++i) lc[i] = lc_g[i];
  const int iters = it_g[0];

  const int tile0  = blockIdx.x * TILES_PER_BLOCK;
  const int lane   = threadIdx.x & 31;
  const int waveid = __builtin_amdgcn_readfirstlane((int)(threadIdx.x >> 5));
  const int myTiles = min(TILES_PER_BLOCK, ntiles_total - tile0);
  const long long total_elems = (long long)rows_total * C_CLS;

  float partial = 0.0f;

  if (myTiles > 0) {
    // Prologue: DMA tile 0 (inputs + targets) into buffer 0. Wave 0 only —
    // scalar branch via readfirstlane so skipped waves never issue the TDM op.
    if (waveid == 0) {
      long long b0 = (long long)tile0 * TILE_ELEMS;
      long long rem = total_elems - b0;
      uint32_t n0 = rem > TILE_ELEMS ? (uint32_t)TILE_ELEMS : (uint32_t)rem;
      tdm_load_1d((uint32_t)(uintptr_t)&xbuf[0][0], xg + b0, n0);
      tdm_load_1d((uint32_t)(uintptr_t)&tbuf[0][0], tg + b0, n0);
    }

    for (int i = 0; i < myTiles; ++i) {
      const int buf = i & 1;
      if (waveid == 0) {
        if (i + 1 < myTiles) {
          // Kick off next tile's DMA, then wait for the current tile's two
          // loads (TDM ops complete in order within a wave).
          long long bn = (long long)(tile0 + i + 1) * TILE_ELEMS;
          long long rem = total_elems - bn;
          uint32_t nn = rem > TILE_ELEMS ? (uint32_t)TILE_ELEMS : (uint32_t)rem;
          tdm_load_1d((uint32_t)(uintptr_t)&xbuf[buf ^ 1][0], xg + bn, nn);
          tdm_load_1d((uint32_t)(uintptr_t)&tbuf[buf ^ 1][0], tg + bn, nn);
          __builtin_amdgcn_s_wait_tensorcnt(2);
        } else {
          __builtin_amdgcn_s_wait_tensorcnt(0);
        }
      }
      __syncthreads();  // tile i now visible in LDS to all waves

      const int row  = threadIdx.x;
      const int grow = (tile0 + i) * ROWS_PER_TILE + row;
      if (row < ROWS_PER_TILE && grow < rows_total) {
        const float* xr = &xbuf[buf][row * C_CLS];
        const float* tr = &tbuf[buf][row * C_CLS];
        float2 xp[C_CLS / 2], tp[C_CLS / 2], ep[C_CLS / 2];
#pragma unroll
        for (int p = 0; p < C_CLS / 2; ++p) {        // 8B-aligned ds_load_b64s
          xp[p] = ((const float2*)xr)[p];
          tp[p] = ((const float2*)tr)[p];
        }
        // e = polyexp(x)  (Horner, pairwise)
        float s = 0.0f;
#pragma unroll
        for (int p = 0; p < C_CLS / 2; ++p) {
          float2 r = f2_bcast(ec[NEXP - 1]);
#pragma unroll
          for (int k = NEXP - 2; k >= 0; --k) r = f2_fma(r, xp[p], f2_bcast(ec[k]));
          ep[p] = r;
          s += r.x + r.y;
        }
        // inv = polyinv(s) refined by Newton-Raphson
        float invv = ic[NINV - 1];
#pragma unroll
        for (int k = NINV - 2; k >= 0; --k) invv = fmaf(invv, s, ic[k]);
        for (int k = 0; k < iters; ++k) invv = invv * fmaf(-s, invv, 2.0f);
        // acc = sum_j t_j * polylog(e_j * inv)
        float acc = 0.0f;
        const float2 iv2 = f2_bcast(invv);
#pragma unroll
        for (int p = 0; p < C_CLS / 2; ++p) {
          float2 sm = f2_mul(ep[p], iv2);
          float2 r = f2_bcast(lc[NLOG - 1]);
#pragma unroll
          for (int k = NLOG - 2; k >= 0; --k) r = f2_fma(r, sm, f2_bcast(lc[k]));
          acc = fmaf(tp[p].x, r.x, acc);
          acc = fmaf(tp[p].y, r.y, acc);
        }
        partial += acc;
      }
      __syncthreads();  // buffer reuse safe for next DMA
    }
  }

  // Deterministic block reduction: wave32 shuffle tree, then LDS across waves.
#pragma unroll
  for (int off = 16; off > 0; off >>= 1) partial += __shfl_down(partial, off, 32);
  if (lane == 0) wred[threadIdx.x >> 5] = partial;
  __syncthreads();
  if (threadIdx.x == 0) {
    float tot = 0.0f;
#pragma unroll
    for (int w = 0; w < NT_MAIN / 32; ++w) tot += wred[w];
    partials[blockIdx.x] = tot;  // every launched block writes its slot
  }
}

__global__ __launch_bounds__(512)
void ce_poly_reduce(const float* __restrict__ partials, int n,
                    float* __restrict__ out, float scale) {
  __shared__ float wred[16];
  float s = 0.0f;
  for (int i = threadIdx.x; i < n; i += 512) s += partials[i];
#pragma unroll
  for (int off = 16; off > 0; off >>= 1) s += __shfl_down(s, off, 32);
  const int lane = threadIdx.x & 31;
  if (lane == 0) wred[threadIdx.x >> 5] = s;
  __syncthreads();
  if (threadIdx.x == 0) {
    float tot = 0.0f;
#pragma unroll
    for (int w = 0; w < 16; ++w) tot += wred[w];
    out[0] = tot * scale;   // scale = -1/B
  }
}

extern "C" void kernel_launch(void* const* d_in, const int* in_sizes, int n_in,
                              void* d_out, int out_size, void* d_ws, size_t ws_size,
                              hipStream_t stream) {
  (void)n_in; (void)out_size; (void)ws_size;
  const float* xg = (const float*)d_in[0];   // enc_input  [B, 10] f32
  const float* tg = (const float*)d_in[1];   // enc_target [B, 10] f32
  const float* ec = (const float*)d_in[2];   // exp coeffs (9)
  const float* ic = (const float*)d_in[3];   // inverse coeffs (5)
  const float* lc = (const float*)d_in[4];   // log coeffs (9)
  const int*   it = (const int*)d_in[5];     // iterations

  const int rows   = in_sizes[0] / C_CLS;                        // B
  const int ntiles = (rows + ROWS_PER_TILE - 1) / ROWS_PER_TILE; // 8000
  const int nb     = (ntiles + TILES_PER_BLOCK - 1) / TILES_PER_BLOCK; // 2000

  float* partials = (float*)d_ws;  // nb floats of scratch (8 KB)

  ce_poly_main<<<nb, NT_MAIN, 0, stream>>>(xg, tg, ec, ic, lc, it,
                                           partials, rows, ntiles);
  ce_poly_reduce<<<1, 512, 0, stream>>>(partials, nb, (float*)d_out,
                                        -1.0f / (float)rows);
}